// FitTorch_53283364274307
// MI455X (gfx1250) — compile-verified
//
#include <hip/hip_runtime.h>

#define N_ATOMS   10000
#define K_NEIGH   32
#define P_PAIRS   (N_ATOMS * K_NEIGH)
#define NUM_RADIAL 8
#define NUM_3BODY  8
#define HIDDEN     64
#define CUTOFF     5.0f
#define ETA        2.0f
#define N_STRUCT   5
#define DESC_W     16
#define PI_F       3.14159265358979323846f

typedef float v2f __attribute__((ext_vector_type(2)));
typedef float v8f __attribute__((ext_vector_type(8)));

static __device__ __forceinline__ v8f wmma4(v2f a, v2f b, v8f c) {
  // D = A(16x4,f32) * B(4x16,f32) + C(16x16,f32)
  return __builtin_amdgcn_wmma_f32_16x16x4_f32(false, a, false, b, (short)0, c,
                                               false, false);
}

// Single-instruction v_rcp_f32 (~1 ulp) instead of the 7-op IEEE div sequence.
static __device__ __forceinline__ float fast_rcp(float x) {
  return __builtin_amdgcn_rcpf(x);
}

// Branchless tanh: 1 - 2/(exp(2x)+1). Straight-line v_exp_f32 + v_rcp_f32,
// no exec-mask divergence (library tanhf emits saveexec branches per call).
static __device__ __forceinline__ float fast_tanh(float x) {
  float t = __expf(2.0f * x);
  return 1.0f - 2.0f * fast_rcp(t + 1.0f);
}

// ---------------------------------------------------------------- K0: zero
__global__ void zero_kernel(float* __restrict__ p, int n) {
  int i = blockIdx.x * blockDim.x + threadIdx.x;
  if (i < n) p[i] = 0.0f;
}

// -------------------------------------------- K1: geometry + radial basis
__global__ void geom_rbf_kernel(const float* __restrict__ x,
                                const float* __restrict__ tx,
                                const int* __restrict__ ui,
                                const int* __restrict__ uj,
                                float* __restrict__ desc,
                                float4* __restrict__ geom) {
  int p = blockIdx.x * blockDim.x + threadIdx.x;   // exact grid, no guard
  int i = ui[p], j = uj[p];
  float dx = x[3 * i + 0] - tx[3 * p + 0] - x[3 * j + 0];
  float dy = x[3 * i + 1] - tx[3 * p + 1] - x[3 * j + 1];
  float dz = x[3 * i + 2] - tx[3 * p + 2] - x[3 * j + 2];
  float dij = sqrtf(dx * dx + dy * dy + dz * dz);
  float invn = fast_rcp(fmaxf(dij, 1e-12f));       // F.normalize clamp
  geom[p] = make_float4(dx * invn, dy * invn, dz * invn, dij);
  const float s2c = 0.63245553203367586f;          // sqrt(2/CUTOFF)
  float invd = fast_rcp(dij);                      // reference uses raw dij
  float wd = (PI_F / CUTOFF) * dij;
  float* drow = desc + (long)p * DESC_W;
#pragma unroll
  for (int n = 1; n <= NUM_RADIAL; ++n)
    drow[n - 1] = s2c * __sinf((float)n * wd) * invd;
}

// -------------------- K2: WMMA MLP forward + analytic backward + atomics
__global__ void mlp_pair_kernel(const float* __restrict__ desc,
                                const float4* __restrict__ geom,
                                const float* __restrict__ W1,
                                const float* __restrict__ b1,
                                const float* __restrict__ W2,
                                const float* __restrict__ b2,
                                const int* __restrict__ ui,
                                const int* __restrict__ indices,
                                const int* __restrict__ uj,
                                float* __restrict__ energy,
                                float* __restrict__ forces) {
  __shared__ float sW1[NUM_RADIAL * HIDDEN];
  __shared__ float sb1[HIDDEN];
  __shared__ float sW2[HIDDEN];
  __shared__ float stage[8][16][2];                 // per-wave {net, dnet/dd}

  int tid = threadIdx.x;
  for (int t = tid; t < NUM_RADIAL * HIDDEN; t += 256) sW1[t] = W1[t];
  if (tid < HIDDEN) { sb1[tid] = b1[tid]; sW2[tid] = W2[tid]; }
  __syncthreads();

  const int wave = tid >> 5;
  const int lane = tid & 31;
  const int m    = lane & 15;                       // row (pair) / col (h) idx
  const int half = lane >> 4;
  const int p0   = (blockIdx.x * 8 + wave) * 16;    // 16 pairs per wave

  float4 geo = geom[p0 + m];
  float dij  = geo.w;

  // --- A tiles: rbf[16x8] split into two K=4 chunks (ISA 16x4 f32 layout)
  const int kb0 = 2 * half;
  const int kb1 = 4 + 2 * half;
  const float* drow = desc + (long)(p0 + m) * DESC_W;
  v2f a0; a0.x = drow[kb0]; a0.y = drow[kb0 + 1];
  v2f a1; a1.x = drow[kb1]; a1.y = drow[kb1 + 1];

  // --- B tiles: W1 chunks, 4 hidden-column tiles of 16
  v2f b0t[4], b1t[4];
#pragma unroll
  for (int t = 0; t < 4; ++t) {
    int n = 16 * t + m;
    b0t[t].x = sW1[(kb0    ) * HIDDEN + n];
    b0t[t].y = sW1[(kb0 + 1) * HIDDEN + n];
    b1t[t].x = sW1[(kb1    ) * HIDDEN + n];
    b1t[t].y = sW1[(kb1 + 1) * HIDDEN + n];
  }

  v8f cz = {0.f, 0.f, 0.f, 0.f, 0.f, 0.f, 0.f, 0.f};
  v8f pre[4];
#pragma unroll
  for (int t = 0; t < 4; ++t) {                     // pre = rbf @ W1
    pre[t] = wmma4(a0, b0t[t], cz);
    pre[t] = wmma4(a1, b1t[t], pre[t]);
  }

  // forward nonlinearity + W2 contraction; keep u = (1-th^2)*W2 for backward
  float po[8];
  v8f u[4];
#pragma unroll
  for (int r = 0; r < 8; ++r) po[r] = 0.f;
#pragma unroll
  for (int t = 0; t < 4; ++t) {
    float bb = sb1[16 * t + m];
    float w2 = sW2[16 * t + m];
#pragma unroll
    for (int r = 0; r < 8; ++r) {
      float th = fast_tanh(pre[t][r] + bb);
      po[r]  += th * w2;
      u[t][r] = (1.f - th * th) * w2;
    }
  }

  // --- JVP GEMM: v = (drbf/dd) @ W1  (same shape/layout as forward)
  float invd = fast_rcp(dij);
  v2f va0, va1;
  {
    const float s2c = 0.63245553203367586f;
    const float wb  = PI_F / CUTOFF;
    float kk[4] = {(float)(kb0 + 1), (float)(kb0 + 2),
                   (float)(kb1 + 1), (float)(kb1 + 2)};
    float dv[4];
#pragma unroll
    for (int q = 0; q < 4; ++q) {
      float w  = kk[q] * wb;
      float sw = __sinf(w * dij);
      float cw = __cosf(w * dij);
      dv[q] = s2c * (w * cw - sw * invd) * invd;    // d(rbf)/d(dij)
    }
    va0.x = dv[0]; va0.y = dv[1]; va1.x = dv[2]; va1.y = dv[3];
  }
  v8f vp[4];
#pragma unroll
  for (int t = 0; t < 4; ++t) {
    vp[t] = wmma4(va0, b0t[t], cz);
    vp[t] = wmma4(va1, b1t[t], vp[t]);
  }
  float sg[8];
#pragma unroll
  for (int r = 0; r < 8; ++r) sg[r] = 0.f;
#pragma unroll
  for (int t = 0; t < 4; ++t)
#pragma unroll
    for (int r = 0; r < 8; ++r) sg[r] += u[t][r] * vp[t][r];

  // reduce over the 16 h-lanes (masks <= 8 stay inside each 16-lane half)
#pragma unroll
  for (int mask = 1; mask <= 8; mask <<= 1)
#pragma unroll
    for (int r = 0; r < 8; ++r) {
      po[r] += __shfl_xor(po[r], mask, 32);
      sg[r] += __shfl_xor(sg[r], mask, 32);
    }

  if (m == 0) {                                     // lanes 0 and 16
#pragma unroll
    for (int r = 0; r < 8; ++r) {
      stage[wave][8 * half + r][0] = po[r];
      stage[wave][8 * half + r][1] = sg[r];
    }
  }
  __syncthreads();

  if (half == 0) {                                  // lanes 0..15: one pair each
    int p = p0 + lane;                              // geo/dij match (m == lane)
    float net = stage[wave][lane][0] + b2[0];
    float s   = stage[wave][lane][1];
    float wd  = (PI_F / CUTOFF) * dij;
    float fc  = 0.5f * (__cosf(wd) + 1.f);
    float dfc = -0.5f * (PI_F / CUTOFF) * __sinf(wd);
    float eij = net * fc;
    float g   = fc * s + net * dfc;                 // d(eij)/d(dij)
    float gx = g * geo.x, gy = g * geo.y, gz = g * geo.z;
    int j = uj[p];
    atomicAdd(&forces[3 * j + 0], gx);              // forces = -grad
    atomicAdd(&forces[3 * j + 1], gy);
    atomicAdd(&forces[3 * j + 2], gz);
    // all 16 pairs share center atom i and structure id -> one atomic each
    float es = eij, ax = gx, ay = gy, az = gz;
#pragma unroll
    for (int mask = 1; mask <= 8; mask <<= 1) {
      es += __shfl_xor(es, mask, 32);
      ax += __shfl_xor(ax, mask, 32);
      ay += __shfl_xor(ay, mask, 32);
      az += __shfl_xor(az, mask, 32);
    }
    if (lane == 0) {
      int i = ui[p0];
      atomicAdd(&forces[3 * i + 0], -ax);
      atomicAdd(&forces[3 * i + 1], -ay);
      atomicAdd(&forces[3 * i + 2], -az);
      atomicAdd(&energy[indices[p0]], es);
    }
  }
}

// ------------------------------- K3: 3-body Gaussian descriptors per atom
__global__ void threebody_kernel(const float4* __restrict__ geom,
                                 float* __restrict__ desc) {
  __shared__ float sdx[32], sdy[32], sdz[32], sfc[32];
  __shared__ float scos[32][33];                    // padded: no bank conflict
  int tid  = threadIdx.x;
  int base = blockIdx.x * K_NEIGH;
  if (tid < 32) {
    float4 g = geom[base + tid];
    sdx[tid] = g.x; sdy[tid] = g.y; sdz[tid] = g.z;
    sfc[tid] = 0.5f * (__cosf((PI_F / CUTOFF) * g.w) + 1.f);
  }
  __syncthreads();
  for (int e = tid; e < 1024; e += 256) {           // cos(theta_kl) matrix
    int k = e >> 5, l = e & 31;
    scos[k][l] = sdx[k] * sdx[l] + sdy[k] * sdy[l] + sdz[k] * sdz[l];
  }
  __syncthreads();
  int k = tid >> 3, mm = tid & 7;                   // thread = (bond k, mu m)
  float mu = -1.f + (2.f / 7.f) * (float)mm;        // linspace(-1,1,8)
  float acc = 0.f;
#pragma unroll
  for (int l = 0; l < K_NEIGH; ++l) {
    float cd = scos[k][l] - mu;
    acc += __expf(-ETA * cd * cd) * sfc[l];
  }
  desc[(long)(base + k) * DESC_W + NUM_RADIAL + mm] = acc;
}

// ------------------------------------------------------------------ launch
extern "C" void kernel_launch(void* const* d_in, const int* in_sizes, int n_in,
                              void* d_out, int out_size, void* d_ws,
                              size_t ws_size, hipStream_t stream) {
  const float* x  = (const float*)d_in[0];
  const float* tx = (const float*)d_in[1];
  const float* W1 = (const float*)d_in[2];
  const float* b1 = (const float*)d_in[3];
  const float* W2 = (const float*)d_in[4];
  const float* b2 = (const float*)d_in[5];
  const int* indices = (const int*)d_in[7];
  const int* ui      = (const int*)d_in[10];
  const int* uj      = (const int*)d_in[11];

  float* out    = (float*)d_out;
  float* energy = out;                              // [5]
  float* forces = out + N_STRUCT;                   // [10000*3]
  float* desc   = out + N_STRUCT + 3 * N_ATOMS;     // [320000*16]
  float4* geom  = (float4*)d_ws;                    // {dn.xyz, dij} per pair

  int nz = N_STRUCT + 3 * N_ATOMS;
  zero_kernel<<<(nz + 255) / 256, 256, 0, stream>>>(out, nz);
  geom_rbf_kernel<<<P_PAIRS / 256, 256, 0, stream>>>(x, tx, ui, uj, desc, geom);
  mlp_pair_kernel<<<P_PAIRS / 128, 256, 0, stream>>>(desc, geom, W1, b1, W2,
                                                     b2, ui, indices, uj,
                                                     energy, forces);
  threebody_kernel<<<N_ATOMS, 256, 0, stream>>>(geom, desc);
}